// ReLUTransform_45423574122847
// MI455X (gfx1250) — compile-verified
//
#include <hip/hip_runtime.h>
#include <hip/hip_bf16.h>

// CDNA5 / gfx1250. wave32. WMMA f32 16x16x4 used as the streaming-reduction engine.

typedef float v2f __attribute__((ext_vector_type(2)));
typedef float v8f __attribute__((ext_vector_type(8)));

constexpr int CHUNK = 1024;          // rows per block (i-dimension chunk)
constexpr int WAVES_PER_BLOCK = 8;   // 256 threads
constexpr int COLS_PER_BLOCK = WAVES_PER_BLOCK * 16;  // 128 columns per block

// ---------------------------------------------------------------------------
// Kernel 0: zero the output accumulators (harness poisons d_out with 0xAA).
// ---------------------------------------------------------------------------
__global__ void relutf_init_kernel(float* __restrict__ out, int n2) {
    int i = blockIdx.x * blockDim.x + threadIdx.x;
    if (i < n2) out[i] = 0.0f;
}

// ---------------------------------------------------------------------------
// Kernel 1: streaming masked matvec via WMMA.
//   S_u[j] = sum_i c[i]*Wu[i,j] + r[i]*|Wu[i,j]|   (accumulated into out[0..N))
//   S_l[j] = sum_i c[i]*Wl[i,j] - r[i]*|Wl[i,j]|   (accumulated into out[N..2N))
// with c = (init_ub+init_lb)/2, r = (init_ub-init_lb)/2.
//
// Wave layout for V_WMMA_F32_16X16X4_F32 (A 16x4, B 4x16, C 16x16):
//   A: lanes 0-15 -> K=0 (elem0), K=1 (elem1); lanes 16-31 -> K=2, K=3.
//      All 16 A-rows are broadcast-identical (c or +/-r values), so every
//      C row holds the same 16-column partial sums.
//   B: lanes 0-15 -> rows i+0 (elem0) / i+1 (elem1), cols j0+lane;
//      lanes 16-31 -> rows i+2 / i+3, cols j0+(lane-16).
//   C/D: element 0, lanes 0-15 = (M=0, N=lane) -> the 16 column partials.
// ---------------------------------------------------------------------------
__global__ __launch_bounds__(256)
void relutf_matvec_kernel(const float* __restrict__ Wu,
                          const float* __restrict__ Wl,
                          const float* __restrict__ iub,
                          const float* __restrict__ ilb,
                          float* __restrict__ out, int n) {
    __shared__ float cs[CHUNK];
    __shared__ float rs[CHUNK];

    const int m = blockIdx.z;                 // 0 = upper matrix, 1 = lower
    const float* __restrict__ W = (m == 0) ? Wu : Wl;
    const float rsign = (m == 0) ? 0.5f : -0.5f;  // lower bound subtracts r*|w|
    const int chunk0 = blockIdx.y * CHUNK;

    // Stage c[] and (+/-)r[] for this row-chunk into LDS (coalesced reads).
    for (int t = threadIdx.x; t < CHUNK; t += blockDim.x) {
        float u = iub[chunk0 + t];
        float l = ilb[chunk0 + t];
        cs[t] = 0.5f * (u + l);
        rs[t] = rsign * (u - l);
    }
    __syncthreads();

    const int lane  = threadIdx.x & 31;
    const int wave  = threadIdx.x >> 5;
    const int col   = blockIdx.x * COLS_PER_BLOCK + wave * 16 + (lane & 15);
    const int kbase = (lane >> 4) << 1;       // 0 for lanes 0-15, 2 for 16-31

    v8f acc = {};                              // f32 16x16 accumulator (8 VGPRs)

    const float* wp = W + (size_t)(chunk0 + kbase) * (size_t)n + (size_t)col;
    const size_t row4 = (size_t)4 * (size_t)n;

    #pragma unroll 4
    for (int i = 0; i < CHUNK; i += 4) {
        // B tile: 4 rows x 16 cols of W (256B useful per wave-iteration).
        v2f b;
        b.x = wp[0];                           // row chunk0+i+kbase
        b.y = wp[n];                           // row chunk0+i+kbase+1
        // stream ~32 rows ahead -> global_prefetch_b8
        __builtin_prefetch(wp + (size_t)32 * (size_t)n, 0, 0);

        // A tiles: broadcast c / (+/-)r for the 4 K-values of this step.
        v2f ac, ar;
        ac.x = cs[i + kbase];
        ac.y = cs[i + kbase + 1];
        ar.x = rs[i + kbase];
        ar.y = rs[i + kbase + 1];

        // acc += c * w
        acc = __builtin_amdgcn_wmma_f32_16x16x4_f32(
            false, ac, false, b, (short)0, acc, false, false);

        // acc += (+/-)r * |w|
        v2f babs;
        babs.x = __builtin_fabsf(b.x);
        babs.y = __builtin_fabsf(b.y);
        acc = __builtin_amdgcn_wmma_f32_16x16x4_f32(
            false, ar, false, babs, (short)0, acc, false, false);

        wp += row4;
    }

    // C row M=0 lives in acc element 0 of lanes 0-15 (all rows identical).
    if (lane < 16) {
        atomicAdd(&out[(size_t)m * (size_t)n + (size_t)col], acc[0]);
    }
}

// ---------------------------------------------------------------------------
// Kernel 2: per-column ReLU-relaxation scalars, applied in place.
//   ub[j] = lu*(S_u[j] + bias_u[j]) + intercept_u[j]
//   lb[j] = ll*(S_l[j] + bias_l[j])
// ---------------------------------------------------------------------------
__global__ void relutf_finalize_kernel(const float* __restrict__ bu,
                                       const float* __restrict__ bl,
                                       const float* __restrict__ pub,
                                       const float* __restrict__ plb,
                                       float* __restrict__ out, int n) {
    int j = blockIdx.x * blockDim.x + threadIdx.x;
    if (j >= n) return;

    float pu = pub[j];
    float pl = plb[j];

    float lu, iu, ll;
    if (pu <= 0.0f) {                 // case I: always inactive
        lu = 0.0f; iu = 0.0f; ll = 0.0f;
    } else if (pl >= 0.0f) {          // case II: always active
        lu = 1.0f; iu = 0.0f; ll = 1.0f;
    } else {                          // case III: crossing (pu > 0 > pl)
        float d = pu - pl;            // > 0
        lu = pu / d;
        iu = -(pu * pl) / d;
        ll = (pu > -pl) ? 1.0f : 0.0f;  // min-area heuristic init
    }

    out[j]     = lu * (out[j]     + bu[j]) + iu;
    out[n + j] = ll * (out[n + j] + bl[j]);
}

// ---------------------------------------------------------------------------
// Launch. Inputs in setup_inputs() order:
//   0: prev_upper_weights (N*N)  1: prev_lower_weights (N*N)
//   2: prev_upper_bias (N)       3: prev_lower_bias (N)
//   4: prev_ub (N)               5: prev_lb (N)
//   6: prev_init_ub (N)          7: prev_init_lb (N)
// Output: [2, N] float32 (ub row then lb row).
// ---------------------------------------------------------------------------
extern "C" void kernel_launch(void* const* d_in, const int* in_sizes, int n_in,
                              void* d_out, int out_size, void* d_ws, size_t ws_size,
                              hipStream_t stream) {
    (void)n_in; (void)d_ws; (void)ws_size;

    const float* Wu  = (const float*)d_in[0];
    const float* Wl  = (const float*)d_in[1];
    const float* bu  = (const float*)d_in[2];
    const float* bl  = (const float*)d_in[3];
    const float* pub = (const float*)d_in[4];
    const float* plb = (const float*)d_in[5];
    const float* iub = (const float*)d_in[6];
    const float* ilb = (const float*)d_in[7];
    float* out = (float*)d_out;

    const int n = in_sizes[4];        // 8192

    // Phase 0: zero accumulators.
    relutf_init_kernel<<<(out_size + 255) / 256, 256, 0, stream>>>(out, out_size);

    // Phase 1: streaming WMMA matvec. grid = (j-tiles, i-chunks, 2 matrices).
    dim3 grid(n / COLS_PER_BLOCK, n / CHUNK, 2);
    relutf_matvec_kernel<<<grid, WAVES_PER_BLOCK * 32, 0, stream>>>(
        Wu, Wl, iub, ilb, out, n);

    // Phase 2: apply lambdas / intercepts in place.
    relutf_finalize_kernel<<<(n + 255) / 256, 256, 0, stream>>>(
        bu, bl, pub, plb, out, n);
}